// KGNN_72404558676162
// MI455X (gfx1250) — compile-verified
//
#include <hip/hip_runtime.h>

typedef float v2f __attribute__((ext_vector_type(2)));
typedef float v4f __attribute__((ext_vector_type(4)));
typedef float v8f __attribute__((ext_vector_type(8)));

#define D 64
#define NCLS 16
#define WROW 68   // padded LDS row stride (floats): lane bank step = 4 -> no conflicts

// ---------------------------------------------------------------------------
// V_WMMA_F32_16X16X4_F32 wrapper.  A: 16x4 f32 (2 VGPR/lane), B: 4x16 f32
// (2 VGPR/lane), C/D: 16x16 f32 (8 VGPR/lane).
// ---------------------------------------------------------------------------
__device__ __forceinline__ v8f wmma_f32_k4(v2f a, v2f b, v8f c) {
    return __builtin_amdgcn_wmma_f32_16x16x4_f32(
        /*neg_a=*/false, a, /*neg_b=*/false, b,
        /*c_mod=*/(short)0, c, /*reuse_a=*/false, /*reuse_b=*/false);
}

// ---------------------------------------------------------------------------
// Zero a float buffer (float4 granularity).
// ---------------------------------------------------------------------------
__global__ void KGNN_zero_kernel(float* __restrict__ p, int n4) {
    int i = blockIdx.x * blockDim.x + threadIdx.x;
    if (i < n4) {
        v4f z = {0.f, 0.f, 0.f, 0.f};
        ((v4f*)p)[i] = z;
    }
}

// ---------------------------------------------------------------------------
// Scatter-add aggregation: agg[dst[e]] += feat[src[e]]  (segment_sum).
// 16 threads per edge, each handles 4 contiguous floats via
// global_atomic_add_f32.
// ---------------------------------------------------------------------------
__global__ void KGNN_scatter_kernel(const float* __restrict__ feat,
                                    const long long* __restrict__ src,
                                    const long long* __restrict__ dst,
                                    float* __restrict__ agg, int E) {
    int t = blockIdx.x * blockDim.x + threadIdx.x;
    int e = t >> 4;
    if (e >= E) return;
    int c = (t & 15) << 2;
    long long s = src[e];
    long long d = dst[e];
    v4f v = *(const v4f*)(feat + (size_t)s * D + c);
    float* o = agg + (size_t)d * D + c;
    atomicAdd(o + 0, v.x);
    atomicAdd(o + 1, v.y);
    atomicAdd(o + 2, v.z);
    atomicAdd(o + 3, v.w);
}

// ---------------------------------------------------------------------------
// Fused graph-conv GEMM:  out[m, :] = agg[m,:] @ Wa^T + xr[m,:] @ Wb^T + bias
// Wa, Wb are [64,64] row-major.  Both weight matrices are staged once per
// block into padded LDS (8 waves share them), then each wave computes a
// 16x64 output stripe with V_WMMA_F32_16X16X4_F32, streaming B operands
// from LDS (ds_load_b64, conflict-free via WROW=68 padding) concurrently
// with the vmem feature loads.
// No __restrict__ on out/xr: layer-2 runs in-place (out == xr), safe
// per-wave (all reads of its own 16 rows complete before stores).
// ---------------------------------------------------------------------------
__global__ void KGNN_gemm64_kernel(const float* agg, const float* xr,
                                   const float* __restrict__ Wa,
                                   const float* __restrict__ Wb,
                                   const float* __restrict__ bias,
                                   float* out, int nrows) {
    __shared__ float lW[2][D * WROW];

    // ---- Stage weights to LDS: 64 rows x 16 float4 per matrix ----
    {
        int tid = threadIdx.x;
#pragma unroll
        for (int i = tid; i < D * 16; i += 256) {
            int r  = i >> 4;
            int c4 = (i & 15) << 2;
            *(v4f*)(&lW[0][r * WROW + c4]) = *(const v4f*)(Wa + r * D + c4);
            *(v4f*)(&lW[1][r * WROW + c4]) = *(const v4f*)(Wb + r * D + c4);
        }
    }
    __syncthreads();

    int wave = blockIdx.x * (blockDim.x >> 5) + (threadIdx.x >> 5);
    int m0 = wave * 16;
    if (m0 >= nrows) return;           // wave-uniform: EXEC all-ones for WMMA

    int lane = threadIdx.x & 31;
    int half = lane >> 4;              // 0: lanes 0-15, 1: lanes 16-31
    int l    = lane & 15;

    // Accumulators for the four 16-col tiles; init with broadcast bias.
    v8f acc[4];
#pragma unroll
    for (int nt = 0; nt < 4; ++nt) {
        float bv = bias[nt * 16 + l];
#pragma unroll
        for (int i = 0; i < 8; ++i) acc[nt][i] = bv;
    }

    const size_t arow = (size_t)(m0 + l) * D;
#pragma unroll
    for (int k0 = 0; k0 < D; k0 += 4) {
        int ka = k0 + half * 2;        // A lane layout: K = {ka, ka+1}
        v2f aA = *(const v2f*)(agg + arow + ka);
        v2f aX = *(const v2f*)(xr  + arow + ka);
#pragma unroll
        for (int nt = 0; nt < 4; ++nt) {
            int col = nt * 16 + l;     // B[k][n] = W[col][k] -> contiguous pair
            v2f bA = *(const v2f*)(&lW[0][col * WROW + ka]);
            acc[nt] = wmma_f32_k4(aA, bA, acc[nt]);
            v2f bB = *(const v2f*)(&lW[1][col * WROW + ka]);
            acc[nt] = wmma_f32_k4(aX, bB, acc[nt]);
        }
    }

    // C/D layout: VGPR i, lanes 0-15 -> row m0+i, lanes 16-31 -> row m0+8+i.
    int rbase = m0 + half * 8;
#pragma unroll
    for (int nt = 0; nt < 4; ++nt) {
#pragma unroll
        for (int i = 0; i < 8; ++i) {
            out[(size_t)(rbase + i) * D + nt * 16 + l] = acc[nt][i];
        }
    }
}

// ---------------------------------------------------------------------------
// Classifier head: out[N,16] = h[N,64] @ W_lin^T + b_lin. One 16-col tile;
// W_lin is only 4 KB and L1-resident, so direct global B loads are fine.
// ---------------------------------------------------------------------------
__global__ void KGNN_gemm16_kernel(const float* __restrict__ h,
                                   const float* __restrict__ Wl,
                                   const float* __restrict__ bias,
                                   float* __restrict__ out, int nrows) {
    int wave = blockIdx.x * (blockDim.x >> 5) + (threadIdx.x >> 5);
    int m0 = wave * 16;
    if (m0 >= nrows) return;

    int lane = threadIdx.x & 31;
    int half = lane >> 4;
    int l    = lane & 15;

    v8f acc;
    {
        float bv = bias[l];
#pragma unroll
        for (int i = 0; i < 8; ++i) acc[i] = bv;
    }

    const size_t arow = (size_t)(m0 + l) * D;
#pragma unroll
    for (int k0 = 0; k0 < D; k0 += 4) {
        int ka = k0 + half * 2;
        v2f a = *(const v2f*)(h + arow + ka);
        v2f b = *(const v2f*)(Wl + (size_t)l * D + ka);  // B[k][n]=W_lin[n][k]
        acc = wmma_f32_k4(a, b, acc);
    }

    int rbase = m0 + half * 8;
#pragma unroll
    for (int i = 0; i < 8; ++i) {
        out[(size_t)(rbase + i) * NCLS + l] = acc[i];
    }
}

// ---------------------------------------------------------------------------
// Launch: zero agg -> scatter x -> gemm1 (h) -> zero agg -> scatter h ->
//         gemm2 (h in-place) -> head (d_out).  Scratch: agg + h = 2*N*64 f32.
// ---------------------------------------------------------------------------
extern "C" void kernel_launch(void* const* d_in, const int* in_sizes, int n_in,
                              void* d_out, int out_size, void* d_ws, size_t ws_size,
                              hipStream_t stream) {
    const float*      x      = (const float*)d_in[0];
    const long long*  ei     = (const long long*)d_in[1];   // int64 [2,E]
    const float*      Wrel1  = (const float*)d_in[2];
    const float*      brel1  = (const float*)d_in[3];
    const float*      Wroot1 = (const float*)d_in[4];
    const float*      Wrel2  = (const float*)d_in[5];
    const float*      brel2  = (const float*)d_in[6];
    const float*      Wroot2 = (const float*)d_in[7];
    const float*      Wlin   = (const float*)d_in[8];
    const float*      blin   = (const float*)d_in[9];
    float*            out    = (float*)d_out;

    const int n = in_sizes[0] / D;        // 50000
    const int e = in_sizes[1] / 2;        // 800000
    const long long* src = ei;
    const long long* dst = ei + e;

    float* agg = (float*)d_ws;
    float* h   = agg + (size_t)n * D;

    const int n4         = n * D / 4;
    const int zero_grid  = (n4 + 255) / 256;
    const int scat_grid  = (e * 16 + 255) / 256;
    const int tiles      = (n + 15) / 16;
    const int gemm_grid  = (tiles + 7) / 8;   // 8 waves (256 thr) per block

    // ---- Layer 1 ----
    KGNN_zero_kernel<<<zero_grid, 256, 0, stream>>>(agg, n4);
    KGNN_scatter_kernel<<<scat_grid, 256, 0, stream>>>(x, src, dst, agg, e);
    KGNN_gemm64_kernel<<<gemm_grid, 256, 0, stream>>>(agg, x, Wrel1, Wroot1,
                                                      brel1, h, n);
    // ---- Layer 2 ----
    KGNN_zero_kernel<<<zero_grid, 256, 0, stream>>>(agg, n4);
    KGNN_scatter_kernel<<<scat_grid, 256, 0, stream>>>(h, src, dst, agg, e);
    KGNN_gemm64_kernel<<<gemm_grid, 256, 0, stream>>>(agg, h, Wrel2, Wroot2,
                                                      brel2, h, n);   // in-place
    // ---- Head ----
    KGNN_gemm16_kernel<<<gemm_grid, 256, 0, stream>>>(h, Wlin, blin, out, n);
}